// IOU_18476949307730
// MI455X (gfx1250) — compile-verified
//
#include <hip/hip_runtime.h>
#include <hip/hip_bf16.h>

typedef __attribute__((ext_vector_type(2))) float v2f;
typedef __attribute__((ext_vector_type(8))) float v8f;

#define PI_F 3.14159265358979323846f

// xor-lane shuffle via DS_SWIZZLE_B32 group-of-32 mode:
// lane' = ((lane & 0x1f) | 0) ^ XORMASK  -> offset = (XORMASK<<10) | 0x1f
template <int XORMASK>
__device__ __forceinline__ float swz_xor(float v) {
    return __int_as_float(
        __builtin_amdgcn_ds_swizzle(__float_as_int(v), (XORMASK << 10) | 0x1f));
}

// Branchless dynamic-index emit: compiles to v_cndmask chains, keeps the
// polygon fully register-resident (no scratch).
template <int CAP>
__device__ __forceinline__ void emit_vertex(float (&Ox)[CAP], float (&Oy)[CAP],
                                            int& cnt, bool pred, float vx, float vy) {
#pragma unroll
    for (int s = 0; s < CAP; ++s) {
        bool wsel = pred && (s == cnt);
        Ox[s] = wsel ? vx : Ox[s];
        Oy[s] = wsel ? vy : Oy[s];
    }
    cnt += pred ? 1 : 0;
}

// One Sutherland-Hodgman half-plane clip. Convexity bounds the live polygon:
// at most CAP_IN vertices in, at most CAP_IN+1 == CAP_OUT out.
template <int CAP_IN, int CAP_OUT>
__device__ __forceinline__ int clip_edge(float (&Px)[8], float (&Py)[8], int n,
                                         float ax, float ay, float bx, float by) {
    float ex = bx - ax, ey = by - ay;

    float sv[CAP_IN];
#pragma unroll
    for (int k = 0; k < CAP_IN; ++k)
        sv[k] = ex * (Py[k] - ay) - ey * (Px[k] - ax);

    float Ox[CAP_OUT], Oy[CAP_OUT];
#pragma unroll
    for (int k = 0; k < CAP_OUT; ++k) { Ox[k] = 0.f; Oy[k] = 0.f; }
    int cnt = 0;

#pragma unroll
    for (int k = 0; k < CAP_IN; ++k) {
        float nx, ny, snx;
        if (k < CAP_IN - 1) {
            bool hn = (k + 1 < n);
            nx = hn ? Px[k + 1] : Px[0];
            ny = hn ? Py[k + 1] : Py[0];
            snx = hn ? sv[k + 1] : sv[0];
        } else {  // k+1 == CAP_IN >= n: always wraps
            nx = Px[0]; ny = Py[0]; snx = sv[0];
        }
        float cx = Px[k], cy = Py[k];
        float sc = sv[k];
        bool active = (k < n);
        float denom = sc - snx;
        float dd = (fabsf(denom) < 1e-9f) ? 1.0f : denom;
        float t = sc / dd;
        float ipx = cx + t * (nx - cx);
        float ipy = cy + t * (ny - cy);
        bool inCur = (sc >= 0.f);
        bool inNxt = (snx >= 0.f);
        emit_vertex<CAP_OUT>(Ox, Oy, cnt, active && inCur, cx, cy);
        emit_vertex<CAP_OUT>(Ox, Oy, cnt, active && (inCur != inNxt), ipx, ipy);
    }
#pragma unroll
    for (int k = 0; k < CAP_OUT; ++k) { Px[k] = Ox[k]; Py[k] = Oy[k]; }
    return cnt;
}

__device__ __forceinline__ void make_corners(float x, float y, float h, float w,
                                             float s, float c,
                                             float (&Qx)[4], float (&Qy)[4]) {
    float hx = 0.5f * w, hy = 0.5f * h;
    float lx[4] = {-hx, hx, hx, -hx};
    float ly[4] = {-hy, -hy, hy, hy};
#pragma unroll
    for (int k = 0; k < 4; ++k) {
        Qx[k] = c * lx[k] - s * ly[k] + x;
        Qy[k] = s * lx[k] + c * ly[k] + y;
    }
}

__device__ __forceinline__ float quad_area4(const float (&Qx)[4], const float (&Qy)[4]) {
    float a = 0.f;
#pragma unroll
    for (int k = 0; k < 4; ++k) {
        int j = (k + 1) & 3;
        a += Qx[k] * Qy[j] - Qy[k] * Qx[j];
    }
    return 0.5f * fabsf(a);
}

__device__ __forceinline__ float quad_intersection(const float (&Q1x)[4], const float (&Q1y)[4],
                                                   const float (&Q2x)[4], const float (&Q2y)[4]) {
    float Px[8], Py[8];
#pragma unroll
    for (int k = 0; k < 8; ++k) { Px[k] = 0.f; Py[k] = 0.f; }
#pragma unroll
    for (int k = 0; k < 4; ++k) { Px[k] = Q1x[k]; Py[k] = Q1y[k]; }

    // Polygon grows by at most one vertex per convex clip: 4->5->6->7->8.
    int n;
    n = clip_edge<4, 5>(Px, Py, 4, Q2x[0], Q2y[0], Q2x[1], Q2y[1]);
    n = clip_edge<5, 6>(Px, Py, n, Q2x[1], Q2y[1], Q2x[2], Q2y[2]);
    n = clip_edge<6, 7>(Px, Py, n, Q2x[2], Q2y[2], Q2x[3], Q2y[3]);
    n = clip_edge<7, 8>(Px, Py, n, Q2x[3], Q2y[3], Q2x[0], Q2y[0]);

    float acc = 0.f;
#pragma unroll
    for (int k = 0; k < 8; ++k) {
        float nx, ny;
        if (k < 7) {
            bool hn = (k + 1 < n);
            nx = hn ? Px[k + 1] : Px[0];
            ny = hn ? Py[k + 1] : Py[0];
        } else {
            nx = Px[0]; ny = Py[0];
        }
        float cross = Px[k] * ny - Py[k] * nx;
        acc += (k < n) ? cross : 0.f;
    }
    return 0.5f * fabsf(acc);
}

__global__ void __launch_bounds__(256)
iou_init_kernel(float* ws) {
    ws[0] = 0.f;
    ws[1] = 0.f;
    reinterpret_cast<unsigned int*>(ws)[2] = 0u;
}

__global__ void __launch_bounds__(256)
iou_main_kernel(const float* __restrict__ inputs_, const float* __restrict__ targets,
                float* __restrict__ ws, int B) {
    int gid = blockIdx.x * blockDim.x + threadIdx.x;
    bool valid = gid < B;
    int i = valid ? gid : 0;   // clamp instead of branching: EXEC stays all-1s for WMMA

    float4 inp = reinterpret_cast<const float4*>(inputs_)[i];
    const float2* t2 = reinterpret_cast<const float2*>(targets + (size_t)6 * (size_t)i);
    float2 ta = t2[0], tb = t2[1], tcv = t2[2];
    float tx = ta.x, ty = ta.y, th = tb.x, tw = tb.y, tsn = tcv.x, tcs = tcv.y;

    float x = inp.x * 500.f + 500.f;
    float y = inp.y * 500.f + 500.f;
    float h = th;                       // reference copies targets' h
    float w = inp.z * 50.f + 50.f;
    float ang = inp.w * PI_F + PI_F;
    float sn = sinf(ang);
    float cs = cosf(ang);

    // loss_i = ||inputs - targets||_2  (h component cancels exactly)
    float dx = x - tx, dy = y - ty, dw = w - tw, dsn = sn - tsn, dcs = cs - tcs;
    float loss = sqrtf(dx * dx + dy * dy + dw * dw + dsn * dsn + dcs * dcs);

    float Q1x[4], Q1y[4], Q2x[4], Q2y[4];
    make_corners(x, y, h, w, sn, cs, Q1x, Q1y);
    make_corners(tx, ty, th, tw, tsn, tcs, Q2x, Q2y);

    float a1 = quad_area4(Q1x, Q1y);
    float a2 = quad_area4(Q2x, Q2y);
    float inter = quad_intersection(Q1x, Q1y, Q2x, Q2y);
    float uni = a1 + a2 - inter;
    float iou = inter / uni;

    loss = valid ? loss : 0.f;
    iou = valid ? iou : 0.f;

    // ---- wave32 reduction stage 1: one V_WMMA_F32_16X16X4_F32 folds the
    // 32 lanes' (loss, iou) pairs halfway AND separates the two sums:
    //   A (16x4, 0/1 selector): rows 0-7 pick K={0,2} (loss), rows 8-15 pick K={1,3} (iou)
    //   B (4x16):  B[0][n]=loss[n], B[2][n]=loss[n+16], B[1][n]=iou[n], B[3][n]=iou[n+16]
    //   D[0][n]   = loss[n]+loss[n+16]   -> D vgpr0, lanes 0-15
    //   D[8][n]   = iou[n]+iou[n+16]     -> D vgpr0, lanes 16-31
    unsigned lane = threadIdx.x & 31u;
    bool iourow = (lane & 8u) != 0u;
    v2f amat, bmat;
    amat.x = iourow ? 0.f : 1.f;   // A[m][0] (lanes 0-15) / A[m][2] (lanes 16-31)
    amat.y = iourow ? 1.f : 0.f;   // A[m][1] / A[m][3]
    bmat.x = loss;                 // B row K=0 / K=2
    bmat.y = iou;                  // B row K=1 / K=3
    v8f cz = {};
    v8f dmat = __builtin_amdgcn_wmma_f32_16x16x4_f32(
        /*neg_a=*/false, amat, /*neg_b=*/false, bmat,
        /*c_mod=*/(short)0, cz, /*reuse_a=*/false, /*reuse_b=*/false);

    // finish both halves' sums in parallel (xor stays within each 16-lane half)
    float v = dmat[0];
    v += swz_xor<1>(v);
    v += swz_xor<2>(v);
    v += swz_xor<4>(v);
    v += swz_xor<8>(v);
    // lanes 0-15 all hold wave sum of loss; lanes 16-31 hold wave sum of iou

    float mx = iou;                          // iou >= 0, so 0-init max is safe
    mx = fmaxf(mx, swz_xor<16>(mx));
    mx = fmaxf(mx, swz_xor<8>(mx));
    mx = fmaxf(mx, swz_xor<4>(mx));
    mx = fmaxf(mx, swz_xor<2>(mx));
    mx = fmaxf(mx, swz_xor<1>(mx));

    __shared__ float sLoss[8], sIou[8], sMax[8];
    unsigned wv = threadIdx.x >> 5;
    if (lane == 0)  { sLoss[wv] = v; sMax[wv] = mx; }
    if (lane == 16) { sIou[wv] = v; }
    __syncthreads();

    if (threadIdx.x == 0) {
        float tl = 0.f, ti = 0.f, tm = 0.f;
#pragma unroll
        for (int k = 0; k < 8; ++k) {
            tl += sLoss[k];
            ti += sIou[k];
            tm = fmaxf(tm, sMax[k]);
        }
        atomicAdd(&ws[0], tl);
        atomicAdd(&ws[1], ti);
        // iou >= 0 -> float ordering == uint ordering on the raw bits
        atomicMax(reinterpret_cast<unsigned int*>(ws) + 2, __float_as_uint(tm));
    }
}

__global__ void __launch_bounds__(256)
iou_finalize_kernel(const float* __restrict__ ws, float* __restrict__ out, int B) {
    float invB = 1.0f / (float)B;
    out[0] = ws[0] * invB;                                          // mean loss
    out[1] = ws[1] * invB;                                          // mean iou
    out[2] = __uint_as_float(reinterpret_cast<const unsigned int*>(ws)[2]);  // max iou
}

extern "C" void kernel_launch(void* const* d_in, const int* in_sizes, int n_in,
                              void* d_out, int out_size, void* d_ws, size_t ws_size,
                              hipStream_t stream) {
    const float* inputs_ = (const float*)d_in[0];   // [B,4] f32
    const float* targets = (const float*)d_in[1];   // [B,6] f32
    int B = in_sizes[0] / 4;
    float* ws = (float*)d_ws;

    iou_init_kernel<<<1, 1, 0, stream>>>(ws);
    int blocks = (B + 255) / 256;
    iou_main_kernel<<<blocks, 256, 0, stream>>>(inputs_, targets, ws, B);
    iou_finalize_kernel<<<1, 1, 0, stream>>>(ws, (float*)d_out, B);
}